// GATMultiHead_4166118277832
// MI455X (gfx1250) — compile-verified
//
#include <hip/hip_runtime.h>
#include <hip/hip_bf16.h>

// Problem constants (match reference)
#define GN     4096   // nodes
#define GC_IN  1024   // input features
#define GH     4      // heads
#define GD     128    // per-head dim (C_OUT = GH*GD = 512)

typedef __attribute__((ext_vector_type(2))) float v2f;
typedef __attribute__((ext_vector_type(8))) float v8f;

// ---------------------------------------------------------------------------
// Kernel A: per-head projection h[hh][n][d] = X[n,:] . W[hh][:,d] + b[hh][d]
// One wave32 computes a full 16x128 strip (16 node rows x all D columns of a
// head) with 8 f32 accumulators, marching K in steps of 4 over C_IN=1024 and
// issuing 8 back-to-back V_WMMA_F32_16X16X4_F32 per K-step. The A operand is
// loaded ONCE per K-step and reused by all 8 column tiles (8x less A traffic
// than one-tile-per-wave).
//
// f32 WMMA operand layouts (ISA 7.12.2):
//   A 16x4:  lanes 0-15 -> M=lane, VGPR0=K0,VGPR1=K1 ; lanes 16-31 -> K2,K3
//   B 4x16:  lanes 0-15 -> N=lane, VGPR0=K0,VGPR1=K1 ; lanes 16-31 -> K2,K3
//   C/D 16x16: VGPR r -> M=r (lanes 0-15) / M=r+8 (lanes 16-31), N=lane&15
// ---------------------------------------------------------------------------
__global__ void __launch_bounds__(32)
gat_proj_wmma(const float* __restrict__ X,   // [N, C_IN]
              const float* __restrict__ W,   // [H, C_IN, D]
              const float* __restrict__ B,   // [H, D]
              float* __restrict__ Hf)        // [H, N, D]
{
    const int nt   = blockIdx.x;           // N tile (0..255)
    const int hh   = blockIdx.y;           // head
    const int lane = threadIdx.x;
    const int half = lane >> 4;            // 0 -> K pair {0,1}, 1 -> {2,3}
    const int l15  = lane & 15;
    const int rowA = nt * 16 + l15;        // node row for A operand

    const float* __restrict__ Wh = W + (size_t)hh * GC_IN * GD;
    const float* __restrict__ Xr = X + (size_t)rowA * GC_IN;

    v8f acc[8] = {};
    for (int k0 = 0; k0 < GC_IN; k0 += 4) {
        const int ka = k0 + half * 2;
        v2f a;
        a.x = Xr[ka];
        a.y = Xr[ka + 1];
        const float* __restrict__ w0 = Wh + (size_t)ka * GD;        // row ka
        const float* __restrict__ w1 = w0 + GD;                     // row ka+1
#pragma unroll
        for (int dt = 0; dt < 8; ++dt) {
            v2f b;
            b.x = w0[dt * 16 + l15];
            b.y = w1[dt * 16 + l15];
            acc[dt] = __builtin_amdgcn_wmma_f32_16x16x4_f32(
                /*neg_a=*/false, a, /*neg_b=*/false, b,
                /*c_mod=*/(short)0, acc[dt], /*reuse_a=*/false, /*reuse_b=*/false);
        }
    }

    const float* __restrict__ bh = B + hh * GD;
    float* __restrict__ outBase = Hf + ((size_t)hh * GN + (size_t)nt * 16) * GD;
#pragma unroll
    for (int dt = 0; dt < 8; ++dt) {
        const int  col  = dt * 16 + l15;
        const float bias = bh[col];
#pragma unroll
        for (int r = 0; r < 8; ++r) {
            const int rowLocal = r + half * 8;           // 0..15 within tile
            outBase[(size_t)rowLocal * GD + col] = acc[dt][r] + bias;
        }
    }
}

// ---------------------------------------------------------------------------
// Kernel B: f1[h][n] = h[h][n][:].v0[h], f2 likewise. One wave32 per (h,n).
// ---------------------------------------------------------------------------
__global__ void __launch_bounds__(256)
gat_attn_coef(const float* __restrict__ Hf,   // [H, N, D]
              const float* __restrict__ v0,   // [H, D]
              const float* __restrict__ v1,   // [H, D]
              float* __restrict__ f1,         // [H, N]
              float* __restrict__ f2)         // [H, N]
{
    const int wave = (int)((blockIdx.x * blockDim.x + threadIdx.x) >> 5); // = h*N+n
    const int lane = threadIdx.x & 31;
    if (wave >= GH * GN) return;
    const int hh = wave / GN;

    const float* __restrict__ hrow = Hf + (size_t)wave * GD;
    float s1 = 0.f, s2 = 0.f;
#pragma unroll
    for (int j = lane; j < GD; j += 32) {
        const float hv = hrow[j];
        s1 += hv * v0[hh * GD + j];
        s2 += hv * v1[hh * GD + j];
    }
#pragma unroll
    for (int off = 16; off > 0; off >>= 1) {
        s1 += __shfl_xor(s1, off, 32);
        s2 += __shfl_xor(s2, off, 32);
    }
    if (lane == 0) { f1[wave] = s1; f2[wave] = s2; }
}

// ---------------------------------------------------------------------------
// Kernel C: one block per (head, row). Pass 1 scans the dense adjacency row
// with float4 (global_load_b128) and compacts nonzeros into LDS with
// vals = sigmoid(adj*(f1+f2)) - 0.5. Block reductions give rowmax/denominator
// for the masked softmax; then each of 128 threads owns one of D=128 output
// columns and accumulates over the compacted neighbor list (coalesced reads
// of h rows).
// ---------------------------------------------------------------------------
#define CAP GN   // worst case: a fully dense row

__global__ void __launch_bounds__(128)
gat_attn_row(const float* __restrict__ adj,  // [N, N]
             const float* __restrict__ Hf,   // [H, N, D]
             const float* __restrict__ f1,   // [H, N]
             const float* __restrict__ f2,   // [H, N]
             float* __restrict__ out)        // [N, H*D]
{
    __shared__ int   s_cnt;
    __shared__ int   s_idx[CAP];
    __shared__ float s_val[CAP];
    __shared__ float s_red[128];

    const int hh  = blockIdx.x;
    const int n   = blockIdx.y;
    const int tid = threadIdx.x;

    if (tid == 0) s_cnt = 0;
    __syncthreads();

    const float  f1n  = f1[hh * GN + n];
    const float4* __restrict__ arow4 = (const float4*)(adj + (size_t)n * GN);
    const float* __restrict__  f2h   = f2 + (size_t)hh * GN;

    // Pass 1: vectorized scan; compact nonzeros, compute sigmoid(logit)-0.5.
    float lmax = -1e30f;
    for (int q = tid; q < GN / 4; q += 128) {
        const float4 av    = arow4[q];
        const int    mbase = q * 4;
        const float  a0[4] = {av.x, av.y, av.z, av.w};
#pragma unroll
        for (int c = 0; c < 4; ++c) {
            const float a = a0[c];
            if (a != 0.0f) {
                const int   m     = mbase + c;
                const float logit = a * (f1n + f2h[m]);
                const float v     = 1.0f / (1.0f + __expf(-logit)) - 0.5f;
                const int   pos   = atomicAdd(&s_cnt, 1);
                s_idx[pos] = m;
                s_val[pos] = v;
                lmax = fmaxf(lmax, v);
            }
        }
    }
    s_red[tid] = lmax;
    __syncthreads();                         // also publishes s_idx/s_val/s_cnt
#pragma unroll
    for (int off = 64; off > 0; off >>= 1) {
        if (tid < off) s_red[tid] = fmaxf(s_red[tid], s_red[tid + off]);
        __syncthreads();
    }
    const float rowmax = s_red[0];
    const int   cnt    = s_cnt;
    __syncthreads();                         // done reading s_red as max

    // Pass 2: exponentiate in place, local partial sum.
    float lsum = 0.f;
    for (int i = tid; i < cnt; i += 128) {
        const float e = __expf(s_val[i] - rowmax);
        s_val[i] = e;
        lsum += e;
    }
    s_red[tid] = lsum;
    __syncthreads();                         // publishes updated s_val too
#pragma unroll
    for (int off = 64; off > 0; off >>= 1) {
        if (tid < off) s_red[tid] += s_red[tid + off];
        __syncthreads();
    }
    const float denom = s_red[0];
    const float inv   = (denom > 0.f) ? (1.0f / fmaxf(denom, 1e-30f)) : 0.f;

    // Pass 3: thread tid owns output column d = tid (D == blockDim.x == 128).
    const float* __restrict__ hbase = Hf + (size_t)hh * GN * GD;
    float acc = 0.f;
    for (int i = 0; i < cnt; ++i) {
        acc += s_val[i] * hbase[(size_t)s_idx[i] * GD + tid];
    }
    // reference output layout: out[n, hh*D + d]
    out[(size_t)n * (GH * GD) + (size_t)hh * GD + tid] = acc * inv;
}

// ---------------------------------------------------------------------------
// Launch
// ---------------------------------------------------------------------------
extern "C" void kernel_launch(void* const* d_in, const int* in_sizes, int n_in,
                              void* d_out, int out_size, void* d_ws, size_t ws_size,
                              hipStream_t stream) {
    const float* node_feats = (const float*)d_in[0];   // [N, C_IN]
    const float* adj        = (const float*)d_in[1];   // [N, N]
    const float* W          = (const float*)d_in[2];   // [H, C_IN, D]
    const float* b          = (const float*)d_in[3];   // [H, D]
    const float* v0         = (const float*)d_in[4];   // [H, D]
    const float* v1         = (const float*)d_in[5];   // [H, D]
    float*       out        = (float*)d_out;           // [N, H*D]

    char*  ws = (char*)d_ws;
    float* Hf = (float*)ws;                                    // H*N*D floats (8 MB)
    float* f1 = (float*)(ws + (size_t)GH * GN * GD * sizeof(float));
    float* f2 = f1 + (size_t)GH * GN;

    // A: projection GEMMs via f32 WMMA (one wave per 16x128 strip)
    gat_proj_wmma<<<dim3(GN / 16, GH), 32, 0, stream>>>(node_feats, W, b, Hf);

    // B: attention coefficients f1/f2 (one wave32 per (h, n))
    {
        const int totalThreads = GH * GN * 32;
        gat_attn_coef<<<totalThreads / 256, 256, 0, stream>>>(Hf, v0, v1, f1, f2);
    }

    // C: sparse masked softmax + aggregation, one block per (head, row)
    gat_attn_row<<<dim3(GH, GN), 128, 0, stream>>>(adj, Hf, f1, f2, out);
}